// _ChamferPure_10256381903339
// MI455X (gfx1250) — compile-verified
//
#include <hip/hip_runtime.h>
#include <hip/hip_bf16.h>
#include <math.h>

typedef __attribute__((ext_vector_type(2))) float v2f;
typedef __attribute__((ext_vector_type(8))) float v8f;

// ---------------------------------------------------------------------------
// Prep: pack each 3-D point into float4 (p0, p1, p2, ||p||^2) in workspace.
// ---------------------------------------------------------------------------
__global__ void pack_points_kernel(const float* __restrict__ x,
                                   const float* __restrict__ y,
                                   float4* __restrict__ xp,
                                   float4* __restrict__ yp, int total) {
  int i = blockIdx.x * blockDim.x + threadIdx.x;
  if (i < total) {
    const float* s = x + (size_t)i * 3;
    float a = s[0], b = s[1], c = s[2];
    xp[i] = make_float4(a, b, c, a * a + b * b + c * c);
    const float* t = y + (size_t)i * 3;
    float d = t[0], e = t[1], f = t[2];
    yp[i] = make_float4(d, e, f, d * d + e * e + f * f);
  }
}

// ---------------------------------------------------------------------------
// Main: one wave owns a 16-row tile of P, streams Q in 16-col tiles.
// Per tile: one v_wmma_f32_16x16x4_f32 computes the full 16x16 squared
// distance tile  D[m][n] = ||p_m||^2 + [-2p_m,1].[q_n,||q_n||^2],
// followed by 8 v_min_num_f32. Clamp-at-0 is deferred past the reduction
// (min and max(.,0) commute).
// B operand = single b64 load per lane from the packed array:
//   lanes 0-15 -> (q0,q1) = K rows 0,1 ; lanes 16-31 -> (q2,||q||^2) = K rows 2,3.
// ---------------------------------------------------------------------------
__global__ __launch_bounds__(256) void chamfer_dir_packed_kernel(
    const float4* __restrict__ P4,  // [B, N] packed row-side points
    const float4* __restrict__ Q4,  // [B, M] packed col-side points
    float* __restrict__ out,        // [B, N] row minima
    int N, int M) {
  const int lane = threadIdx.x & 31;
  const int wave = threadIdx.x >> 5;
  const int half = lane >> 4;  // 0: lanes 0-15, 1: lanes 16-31
  const int l15  = lane & 15;
  const int b    = blockIdx.y;
  const int rowbase = (blockIdx.x * 8 + wave) * 16;  // 8 waves per block

  const float4* __restrict__ P4b = P4 + (size_t)b * N;
  const v2f* __restrict__ Q2 = (const v2f*)(Q4 + (size_t)b * M);

  // A (loop-invariant): row = rowbase + l15.
  // lane<16 holds K=0,1 = (-2p0,-2p1); lane>=16 holds K=2,3 = (-2p2, 1).
  float4 p = P4b[rowbase + l15];
  v2f A;
  A.x = half ? (-2.0f * p.z) : (-2.0f * p.x);
  A.y = half ? 1.0f          : (-2.0f * p.y);

  // C (loop-invariant): C[e] = ||p_{rowbase + e + 8*half}||^2
  v8f C;
#pragma unroll
  for (int e = 0; e < 8; ++e) C[e] = P4b[rowbase + e + 8 * half].w;

  const float inf = __builtin_inff();
  v8f mn = {inf, inf, inf, inf, inf, inf, inf, inf};

#pragma unroll 4
  for (int jt = 0; jt < M; jt += 16) {
    // One b64 load per lane builds the whole B operand.
    v2f Bv = Q2[2 * (jt + l15) + half];
    v8f D = __builtin_amdgcn_wmma_f32_16x16x4_f32(
        false, A, false, Bv, (short)0, C, false, false);
#pragma unroll
    for (int e = 0; e < 8; ++e) mn[e] = fminf(mn[e], D[e]);
  }

  // Min-reduce across the 16 lanes of each half (tile columns).
#pragma unroll
  for (int off = 1; off < 16; off <<= 1) {
#pragma unroll
    for (int e = 0; e < 8; ++e) {
      float o = __shfl_xor(mn[e], off, 16);
      mn[e] = fminf(mn[e], o);
    }
  }

  // lane 0 -> rows rowbase+0..7 ; lane 16 -> rows rowbase+8..15
  if (l15 == 0) {
    float* o = out + (size_t)b * N + rowbase + 8 * half;
#pragma unroll
    for (int e = 0; e < 8; ++e) o[e] = fmaxf(mn[e], 0.0f);  // deferred clamp
  }
}

// ---------------------------------------------------------------------------
// Fallback (no workspace): build operands on the fly from [*,3] layout.
// ---------------------------------------------------------------------------
__global__ __launch_bounds__(256) void chamfer_dir_raw_kernel(
    const float* __restrict__ P, const float* __restrict__ Q,
    float* __restrict__ out, int N, int M) {
  const int lane = threadIdx.x & 31;
  const int wave = threadIdx.x >> 5;
  const int half = lane >> 4;
  const int l15  = lane & 15;
  const int b    = blockIdx.y;
  const int rowbase = (blockIdx.x * 8 + wave) * 16;

  const float* __restrict__ Pb = P + (size_t)b * (size_t)N * 3;
  const float* __restrict__ Qb = Q + (size_t)b * (size_t)M * 3;

  const float* xp = Pb + (size_t)(rowbase + l15) * 3;
  const float x0 = xp[0], x1 = xp[1], x2 = xp[2];
  v2f A;
  A.x = half ? (-2.0f * x2) : (-2.0f * x0);
  A.y = half ? 1.0f         : (-2.0f * x1);

  v8f C;
#pragma unroll
  for (int e = 0; e < 8; ++e) {
    const float* q = Pb + (size_t)(rowbase + e + 8 * half) * 3;
    C[e] = q[0] * q[0] + q[1] * q[1] + q[2] * q[2];
  }

  const float inf = __builtin_inff();
  v8f mn = {inf, inf, inf, inf, inf, inf, inf, inf};

#pragma unroll 4
  for (int jt = 0; jt < M; jt += 16) {
    const float* yp = Qb + (size_t)(jt + l15) * 3;
    const float y0 = yp[0], y1 = yp[1], y2 = yp[2];
    const float ysq = y0 * y0 + y1 * y1 + y2 * y2;
    v2f Bv;
    Bv.x = half ? y2  : y0;
    Bv.y = half ? ysq : y1;
    v8f D = __builtin_amdgcn_wmma_f32_16x16x4_f32(
        false, A, false, Bv, (short)0, C, false, false);
#pragma unroll
    for (int e = 0; e < 8; ++e) mn[e] = fminf(mn[e], D[e]);
  }

#pragma unroll
  for (int off = 1; off < 16; off <<= 1) {
#pragma unroll
    for (int e = 0; e < 8; ++e) {
      float o = __shfl_xor(mn[e], off, 16);
      mn[e] = fminf(mn[e], o);
    }
  }

  if (l15 == 0) {
    float* o = out + (size_t)b * N + rowbase + 8 * half;
#pragma unroll
    for (int e = 0; e < 8; ++e) o[e] = fmaxf(mn[e], 0.0f);
  }
}

extern "C" void kernel_launch(void* const* d_in, const int* in_sizes, int n_in,
                              void* d_out, int out_size, void* d_ws, size_t ws_size,
                              hipStream_t stream) {
  const float* x = (const float*)d_in[0];  // [2, 8192, 3]
  const float* y = (const float*)d_in[1];  // [2, 8192, 3]
  float* out = (float*)d_out;              // [dl (2*8192) | dr (2*8192)]

  const int B = 2, N = 8192, M = 8192;
  const int total = B * N;                 // points per cloud
  dim3 block(256);                         // 8 waves -> 128 rows per block
  dim3 grid(N / 128, B);

  const size_t need = 2 * (size_t)total * sizeof(float4);  // 512 KB
  if (ws_size >= need) {
    float4* xp = (float4*)d_ws;
    float4* yp = xp + total;
    pack_points_kernel<<<(total + 255) / 256, 256, 0, stream>>>(x, y, xp, yp, total);
    // dl: for each x point, min over y
    chamfer_dir_packed_kernel<<<grid, block, 0, stream>>>(xp, yp, out, N, M);
    // dr: for each y point, min over x
    chamfer_dir_packed_kernel<<<grid, block, 0, stream>>>(yp, xp, out + (size_t)B * N, M, N);
  } else {
    chamfer_dir_raw_kernel<<<grid, block, 0, stream>>>(x, y, out, N, M);
    chamfer_dir_raw_kernel<<<grid, block, 0, stream>>>(y, x, out + (size_t)B * N, M, N);
  }
}